// APoTQuantizer_80934363726423
// MI455X (gfx1250) — compile-verified
//
#include <hip/hip_runtime.h>

// APoT fake-quantizer, elementwise over 134M fp32 elements.
// Memory-bound: 1 GiB traffic @ 23.3 TB/s -> ~46 us floor.
// v2: replace per-element 8-step LDS binary search (DS-bound) with a
// per-block 513-entry midpoint LUT holding (left,right) level pairs:
// exactly ONE ds_load_b64 per element + exact distance/tie compare.
// Stream via B128 nontemporal global loads/stores.

typedef float f4 __attribute__((ext_vector_type(4)));
typedef float f2 __attribute__((ext_vector_type(2)));

#define THREADS   256
#define NLEV_PAD  256   // levels padded with +inf -> fixed 8-step search
#define LUT_N     513   // u = floor(x_norm*256)+256 in [0,512]

__global__ void APoTQuantizer_kernel(const float* __restrict__ x,
                                     const float* __restrict__ levels,
                                     const float* __restrict__ raw_alpha,
                                     float* __restrict__ out,
                                     long long n, int nlev) {
    __shared__ float slev[NLEV_PAD];
    __shared__ f2    lut[LUT_N];
    const int tid = threadIdx.x;

    // Stage the level table into LDS, pad with +inf.
    for (int i = tid; i < NLEV_PAD; i += THREADS)
        slev[i] = (i < nlev) ? levels[i] : __builtin_inff();
    __syncthreads();

    // Build the (left,right) candidate-pair LUT. Cell u covers
    // x_norm in [(u-256)/256, (u-255)/256); representative is the cell
    // center, which is exact in fp32 and never equals a level.
    for (int u = tid; u < LUT_N; u += THREADS) {
        float xr = ((float)(u - 256) + 0.5f) * (1.0f / 256.0f);
        int pos = 0;
#pragma unroll
        for (int step = 128; step > 0; step >>= 1) {
            float v = slev[pos + step - 1];
            pos += (v < xr) ? step : 0;      // searchsorted side='left'
        }
        int ridx = (pos < nlev - 1) ? pos : (nlev - 1);
        int lidx = (ridx - 1 > 0) ? (ridx - 1) : 0;
        f2 pr; pr.x = slev[lidx]; pr.y = slev[ridx];
        lut[u] = pr;                          // ds_store_b64
    }
    __syncthreads();

    // alpha = softplus(raw_alpha); uniform -> scalar float pipe.
    float r = raw_alpha[0];
    float alpha = fmaxf(r, 0.0f) + log1pf(expf(-fabsf(r)));
    float inv_alpha = 1.0f / alpha;

    const long long gsize = (long long)gridDim.x * THREADS;
    const long long nvec  = n >> 2;

    // Main stream: one float4 per lane per iteration.
    for (long long i = (long long)blockIdx.x * THREADS + tid; i < nvec; i += gsize) {
        f4 v = __builtin_nontemporal_load((const f4*)x + i);
        f4 o;
#pragma unroll
        for (int c = 0; c < 4; ++c) {
            float xc = fminf(fmaxf(v[c], -alpha), alpha);   // clip(x,-a,a)
            float xn = xc * inv_alpha;                      // /alpha
            int u = (int)floorf(xn * 256.0f) + 256;         // exact in fp32
            u = (u < 0) ? 0 : ((u > 512) ? 512 : u);
            f2 pr = lut[u];                                 // ds_load_b64
            float lv = pr.x, rv = pr.y;
            float nearest = (fabsf(rv - xn) < fabsf(xn - lv)) ? rv : lv; // tie->left
            o[c] = nearest * alpha;
        }
        __builtin_nontemporal_store(o, (f4*)out + i);
    }

    // Scalar tail (n divisible by 4 here, but stay generic).
    for (long long i = (nvec << 2) + (long long)blockIdx.x * THREADS + tid;
         i < n; i += gsize) {
        float xc = fminf(fmaxf(x[i], -alpha), alpha);
        float xn = xc * inv_alpha;
        int u = (int)floorf(xn * 256.0f) + 256;
        u = (u < 0) ? 0 : ((u > 512) ? 512 : u);
        f2 pr = lut[u];
        float lv = pr.x, rv = pr.y;
        out[i] = ((fabsf(rv - xn) < fabsf(xn - lv)) ? rv : lv) * alpha;
    }
}

extern "C" void kernel_launch(void* const* d_in, const int* in_sizes, int n_in,
                              void* d_out, int out_size, void* d_ws, size_t ws_size,
                              hipStream_t stream) {
    (void)n_in; (void)d_ws; (void)ws_size; (void)out_size;
    const float* x      = (const float*)d_in[0];   // (32,2048,2048) fp32
    const float* levels = (const float*)d_in[1];   // sorted, 129 entries
    const float* ralpha = (const float*)d_in[2];   // scalar
    float*       out    = (float*)d_out;

    const long long n    = (long long)in_sizes[0];
    const int       nlev = in_sizes[1];

    long long nvec   = n >> 2;
    long long needed = (nvec + THREADS - 1) / THREADS;
    int blocks = (needed < 8192) ? (int)(needed ? needed : 1) : 8192;

    APoTQuantizer_kernel<<<blocks, THREADS, 0, stream>>>(x, levels, ralpha,
                                                         out, n, nlev);
}